// OrthogonallyAttenuatedLinear_51281909514452
// MI455X (gfx1250) — compile-verified
//
#include <hip/hip_runtime.h>
#include <math.h>

typedef __attribute__((ext_vector_type(2))) float v2f;
typedef __attribute__((ext_vector_type(8))) float v8f;

#define IN_F   128
#define OUT_F  128
#define BATCH  262144

// ws layout (float offsets). Each B-fragment array holds, for every
// (col-tile n in 0..7, k-step k in 0..31, lane in 0..31), the two floats that
// lane needs for V_WMMA_F32_16X16X4_F32's B operand:
//   VGPR0 = M[k*4 + (lane>>4)*2    ][ n*16 + (lane&15) ]
//   VGPR1 = M[k*4 + (lane>>4)*2 + 1][ n*16 + (lane&15) ]
#define WS_BW   0        // W^T fragments       (8192 v2f = 16384 floats)
#define WS_BWN  16384    // wn^T fragments
#define WS_BC   32768    // center^T fragments
#define WS_COFF 49152    // sum_i c_oi * wn_oi   (128)
#define WS_CC   49280    // ||c_o||^2            (128)
#define WS_IS2  49408    // 1 / sigma_o^2        (128)

__global__ __launch_bounds__(128) void oal_prologue(
    const float* __restrict__ W, const float* __restrict__ C,
    const float* __restrict__ sigma, float* __restrict__ ws)
{
  __shared__ float s_inv[OUT_F];
  const int o = threadIdx.x;  // 0..127
  // Row norm of W (F.normalize semantics: clamp norm at 1e-12)
  float ss = 0.f;
  for (int i = 0; i < IN_F; ++i) { float w = W[o * IN_F + i]; ss += w * w; }
  const float nrm = fmaxf(sqrtf(ss), 1e-12f);
  const float inv = 1.0f / nrm;
  s_inv[o] = inv;
  // Per-output scalars
  float coff = 0.f, cc = 0.f;
  for (int i = 0; i < IN_F; ++i) {
    const float c = C[o * IN_F + i];
    coff += c * (W[o * IN_F + i] * inv);
    cc   += c * c;
  }
  ws[WS_COFF + o] = coff;
  ws[WS_CC   + o] = cc;
  const float sg = sigma[o];
  ws[WS_IS2  + o] = 1.0f / (sg * sg);
  __syncthreads();
  // Pre-swizzle B matrices into WMMA fragment layout (8192 v2f entries each)
  for (int e = threadIdx.x; e < 8192; e += 128) {
    const int lane = e & 31;
    const int k    = (e >> 5) & 31;
    const int n    = e >> 10;                  // col tile 0..7
    const int oo   = n * 16 + (lane & 15);     // output column
    const int kk   = k * 4 + ((lane >> 4) << 1);
    const float w0 = W[oo * IN_F + kk];
    const float w1 = W[oo * IN_F + kk + 1];
    const float io = s_inv[oo];
    ws[WS_BW  + 2 * e]     = w0;
    ws[WS_BW  + 2 * e + 1] = w1;
    ws[WS_BWN + 2 * e]     = w0 * io;
    ws[WS_BWN + 2 * e + 1] = w1 * io;
    ws[WS_BC  + 2 * e]     = C[oo * IN_F + kk];
    ws[WS_BC  + 2 * e + 1] = C[oo * IN_F + kk + 1];
  }
}

// 4 waves per WG, each wave owns 16 batch rows; WG covers 64 rows.
// LDS x-tile row stride padded to 132 floats (= 4-bank row stride) so the
// 8-byte A-fragment reads are bank-conflict free. A is re-read from LDS each
// k-step instead of being kept register-resident: keeps VGPR pressure at
// ~3 accumulators + pipeline window, so nothing spills to scratch.
#define TILE_ROWS 64
#define LDS_STRIDE 132

__global__ __launch_bounds__(128) void oal_main(
    const float* __restrict__ x, const float* __restrict__ bias,
    const float* __restrict__ ws, float* __restrict__ out)
{
  __shared__ float tile[TILE_ROWS * LDS_STRIDE];
  const int blockBase = blockIdx.x * TILE_ROWS;

  // Coalesced stage of the 64x128 f32 x-tile into LDS (float4 per thread)
  for (int i = threadIdx.x; i < TILE_ROWS * (IN_F / 4); i += 128) {
    const int rr = i >> 5;            // row in tile
    const int cc = (i & 31) << 2;     // column (float index)
    const float4 v = *reinterpret_cast<const float4*>(
        x + (size_t)(blockBase + rr) * IN_F + cc);
    *reinterpret_cast<float4*>(&tile[rr * LDS_STRIDE + cc]) = v;
  }
  __syncthreads();

  const int lane = threadIdx.x & 31;
  const int wave = threadIdx.x >> 5;
  const int hi16 = lane >> 4;               // which half-wave
  const int lr   = wave * 16 + (lane & 15); // local A row
  const float* arow = &tile[lr * LDS_STRIDE + (hi16 << 1)];  // +4*k per step

  // ||x_row||^2: each half-wave accumulates its half of K, combine with xor-16
  float partial = 0.f;
#pragma unroll
  for (int k = 0; k < 32; ++k) {
    const v2f a = *reinterpret_cast<const v2f*>(arow + k * 4);
    partial += a[0] * a[0] + a[1] * a[1];
  }
  const float xx = partial + __shfl_xor(partial, 16, 32);

  const v2f* bw  = reinterpret_cast<const v2f*>(ws + WS_BW);
  const v2f* bwn = reinterpret_cast<const v2f*>(ws + WS_BWN);
  const v2f* bc  = reinterpret_cast<const v2f*>(ws + WS_BC);

  const int rowBase = blockBase + wave * 16;

  for (int n = 0; n < 8; ++n) {  // 8 output col-tiles of 16
    v8f accW = {0.f, 0.f, 0.f, 0.f, 0.f, 0.f, 0.f, 0.f};
    v8f accN = {0.f, 0.f, 0.f, 0.f, 0.f, 0.f, 0.f, 0.f};
    v8f accC = {0.f, 0.f, 0.f, 0.f, 0.f, 0.f, 0.f, 0.f};
    const v2f* pw = bw  + n * 1024 + lane;
    const v2f* pn = bwn + n * 1024 + lane;
    const v2f* pc = bc  + n * 1024 + lane;
#pragma unroll 4
    for (int k = 0; k < 32; ++k) {
      const v2f av  = *reinterpret_cast<const v2f*>(arow + k * 4);  // LDS
      const v2f bWv = pw[k * 32];                                   // L2-resident
      const v2f bNv = pn[k * 32];
      const v2f bCv = pc[k * 32];
      accW = __builtin_amdgcn_wmma_f32_16x16x4_f32(
          false, av, false, bWv, (short)0, accW, false, false);
      accN = __builtin_amdgcn_wmma_f32_16x16x4_f32(
          false, av, false, bNv, (short)0, accN, false, false);
      accC = __builtin_amdgcn_wmma_f32_16x16x4_f32(
          false, av, false, bCv, (short)0, accC, false, false);
    }
    const int   o    = n * 16 + (lane & 15);
    const float b_o  = bias[o];
    const float coff = ws[WS_COFF + o];
    const float cc_o = ws[WS_CC   + o];
    const float is2  = ws[WS_IS2  + o];
#pragma unroll
    for (int e = 0; e < 8; ++e) {
      // C/D layout: VGPR e holds M = e (lanes 0-15) and M = e+8 (lanes 16-31)
      const int   m    = e + (hi16 << 3);
      const float xxm  = __shfl(xx, m, 32);        // ||x_row m||^2
      const float pwv  = accW[e] + b_o;            // x.W^T + b
      const float pwcv = accN[e] - coff;           // (x-c).wn
      const float xnv  = xxm + cc_o - 2.0f * accC[e];  // ||x-c||^2
      const float poL2 = xnv - pwcv * pwcv;
      const float om   = __expf(-poL2 * is2);
      out[(size_t)(rowBase + m) * OUT_F + o] = pwv * om;
    }
  }
}

extern "C" void kernel_launch(void* const* d_in, const int* in_sizes, int n_in,
                              void* d_out, int out_size, void* d_ws, size_t ws_size,
                              hipStream_t stream) {
  (void)in_sizes; (void)n_in; (void)out_size; (void)ws_size;
  const float* x     = (const float*)d_in[0];
  const float* W     = (const float*)d_in[1];
  const float* bias  = (const float*)d_in[2];
  const float* C     = (const float*)d_in[3];
  const float* sigma = (const float*)d_in[4];
  float* ws  = (float*)d_ws;
  float* out = (float*)d_out;

  oal_prologue<<<1, 128, 0, stream>>>(W, C, sigma, ws);
  oal_main<<<BATCH / TILE_ROWS, 128, 0, stream>>>(x, bias, ws, out);
}